// MSTGCNBlock_68006512165079
// MI455X (gfx1250) — compile-verified
//
#include <hip/hip_runtime.h>

// ---- problem constants (from reference) ----
#define Bn   4
#define Nn   10000
#define FIN  16
#define Tt   12
#define En   160000
#define BT   (Bn*Tt)        // 48
#define RL   (BT*FIN)       // 768 floats per node row of v
#define CC   64             // C_CHEB == C_TIME
#define TOTV (Nn*RL)        // 7,680,000 elements
#define KT   208            // 192 (im2col of 64ch x 3 taps) + 16 (residual feats)
#define AST  210            // LDS stride for im2col panel (even, 210%64=18)
#define LNEPS 1e-5f

typedef float v2f __attribute__((ext_vector_type(2)));
typedef float v8f __attribute__((ext_vector_type(8)));

// ---------------- elementwise / setup kernels ----------------

__global__ void fill0(float* p, int n) {
    int i = blockIdx.x * blockDim.x + threadIdx.x;
    if (i < n) p[i] = 0.0f;
}

__global__ void deg_count(const int* __restrict__ ei, float* __restrict__ deg) {
    int e = blockIdx.x * blockDim.x + threadIdx.x;
    if (e < En) atomicAdd(&deg[ei[e]], 1.0f);   // row = ei[0][e]
}

// x (B,N,F,T) -> v (N, B*T, F)
__global__ void build_v(const float* __restrict__ x, float* __restrict__ v) {
    int i = blockIdx.x * blockDim.x + threadIdx.x;
    if (i >= TOTV) return;
    int t = i % Tt;
    int f = (i / Tt) % FIN;
    int n = (i / (Tt * FIN)) % Nn;
    int b = i / (Tt * FIN * Nn);
    v[(n * BT + b * Tt + t) * FIN + f] = x[i];
}

// agg[col[e]][:] += u[row[e]][:]
__global__ void scatter_edges(const float* __restrict__ u,
                              const int* __restrict__ ei,
                              float* __restrict__ agg) {
    int i = blockIdx.x * blockDim.x + threadIdx.x;
    if (i >= En * RL) return;
    int e = i / RL;
    int j = i - e * RL;
    int r = ei[e];        // row
    int c = ei[En + e];   // col
    atomicAdd(&agg[c * RL + j], u[r * RL + j]);
}

// t1 = (2/lam)*(deg*v - agg) - v
__global__ void combine1(const float* __restrict__ v, const float* __restrict__ agg,
                         const float* __restrict__ deg, const float* __restrict__ lam,
                         float* __restrict__ t1) {
    int i = blockIdx.x * blockDim.x + threadIdx.x;
    if (i >= TOTV) return;
    float cs = 2.0f / lam[0];
    int n = i / RL;
    float vv = v[i];
    t1[i] = cs * (deg[n] * vv - agg[i]) - vv;
}

// agg <- Tx2 = 2*((2/lam)*(deg*t1 - agg) - t1) - v   (in place over agg)
__global__ void combine2(const float* __restrict__ v, const float* __restrict__ t1,
                         const float* __restrict__ deg, const float* __restrict__ lam,
                         float* __restrict__ agg) {
    int i = blockIdx.x * blockDim.x + threadIdx.x;
    if (i >= TOTV) return;
    float cs = 2.0f / lam[0];
    int n = i / RL;
    float u = t1[i];
    float l = cs * (deg[n] * u - agg[i]) - u;
    agg[i] = 2.0f * l - v[i];
}

// ---------------- fused per-node kernel ----------------
// Phase 1: Cheb projection (WMMA f32 16x16x4) -> relu -> LDS sp[48][64]
// Phase 2: im2col into Aim[48][208] (+ residual features), zero-padded per batch
// Phase 3: temporal conv + residual as one WMMA GEMM (K=208) -> relu -> hh
// Phase 4: LayerNorm over channels -> output (B,N,C,T)
__global__ __launch_bounds__(128)
void fused_tail(const float* __restrict__ v, const float* __restrict__ t1,
                const float* __restrict__ t2,
                const float* __restrict__ chebW, const float* __restrict__ chebB,
                const float* __restrict__ x,
                const float* __restrict__ timeW, const float* __restrict__ timeB,
                const float* __restrict__ resW,  const float* __restrict__ resB,
                const float* __restrict__ gamma, const float* __restrict__ beta,
                float* __restrict__ out) {
    // spbuf serves as sp (stride 64) in phases 1-2, then as hh (stride 65) in 3-4.
    __shared__ float spbuf[BT * 65];
    __shared__ float Aim[BT * AST];
    __shared__ float mu[BT];
    __shared__ float rs[BT];

    const int n    = blockIdx.x;
    const int tid  = threadIdx.x;
    const int wave = tid >> 5;
    const int lane = tid & 31;
    const int lm   = lane & 15;
    const int hi2  = (lane >> 4) << 1;   // 0 for lanes 0-15, 2 for lanes 16-31

    // residual features straight into Aim columns [192,208): Aim[bt][192+f] = x(b,n,f,t)
    for (int i = tid; i < BT * FIN; i += 128) {
        int b  = i / (FIN * Tt);
        int rr = i - b * (FIN * Tt);
        int f  = rr / Tt;
        int t  = rr - f * Tt;
        Aim[(b * Tt + t) * AST + 192 + f] = x[((b * Nn + n) * FIN + f) * Tt + t];
    }

    // ---- Phase 1: Cheb WMMA ----
    const float* mats[3] = { v, t1, t2 };
    #pragma unroll
    for (int tno = 0; tno < 3; ++tno) {
        int tile = wave * 3 + tno;         // 0..11
        int rt = tile >> 2;                // row tile 0..2
        int ct = tile & 3;                 // col tile 0..3
        int rowbase = n * BT + rt * 16;
        int c0 = ct * 16;
        v8f acc = {};
        #pragma unroll
        for (int km = 0; km < 3; ++km) {
            const float* U = mats[km];
            const float* W = chebW + km * FIN * CC;
            #pragma unroll
            for (int kb = 0; kb < FIN; kb += 4) {
                int k0 = kb + hi2;
                v2f a = *(const v2f*)(U + (rowbase + lm) * FIN + k0);
                v2f bb;
                bb.x = W[k0 * CC + c0 + lm];
                bb.y = W[(k0 + 1) * CC + c0 + lm];
                acc = __builtin_amdgcn_wmma_f32_16x16x4_f32(
                        false, a, false, bb, (short)0, acc, false, false);
            }
        }
        float bias = chebB[c0 + lm];
        #pragma unroll
        for (int j = 0; j < 8; ++j) {
            int m = (lane < 16) ? j : (j + 8);
            float val = acc[j] + bias;
            spbuf[(rt * 16 + m) * CC + c0 + lm] = val > 0.f ? val : 0.f;  // sp, stride 64
        }
    }
    __syncthreads();

    // ---- Phase 2: im2col k<192: Aim[bt][c*3+dt] = sp[bt+dt-1][c] (0 at T edges) ----
    for (int i = tid; i < BT * 192; i += 128) {
        int bt = i / 192;
        int k  = i - bt * 192;
        int c  = k / 3;
        int dt = k - c * 3;
        int b  = bt / Tt;
        int t  = bt - b * Tt;
        int tp = t + dt - 1;
        float val = (tp >= 0 && tp < Tt) ? spbuf[(bt + dt - 1) * CC + c] : 0.f;
        Aim[bt * AST + k] = val;
    }
    __syncthreads();

    // ---- Phase 3: temporal + residual GEMM via WMMA, K = 208 ----
    #pragma unroll
    for (int tno = 0; tno < 3; ++tno) {
        int tile = wave * 3 + tno;
        int rt = tile >> 2;
        int ct = tile & 3;
        int c0 = ct * 16;
        const float* Arow = Aim + (rt * 16 + lm) * AST;
        const float* Wt   = timeW + (c0 + lm) * 192;   // B[k][o] = timeW[o*192 + k]
        const float* Wr   = resW  + (c0 + lm) * FIN;   // B[192+f][o] = resW[o*16 + f]
        v8f acc = {};
        #pragma unroll 4
        for (int kb = 0; kb < 192; kb += 4) {
            int k0 = kb + hi2;
            v2f a = *(const v2f*)(Arow + k0);          // ds_load_b64
            v2f bb;
            bb.x = Wt[k0];
            bb.y = Wt[k0 + 1];
            acc = __builtin_amdgcn_wmma_f32_16x16x4_f32(
                    false, a, false, bb, (short)0, acc, false, false);
        }
        #pragma unroll
        for (int kb = 192; kb < KT; kb += 4) {
            int k0 = kb + hi2;
            v2f a = *(const v2f*)(Arow + k0);
            v2f bb;
            bb.x = Wr[k0 - 192];
            bb.y = Wr[k0 - 191];
            acc = __builtin_amdgcn_wmma_f32_16x16x4_f32(
                    false, a, false, bb, (short)0, acc, false, false);
        }
        float bias = timeB[c0 + lm] + resB[c0 + lm];
        #pragma unroll
        for (int j = 0; j < 8; ++j) {
            int m = (lane < 16) ? j : (j + 8);
            float hv = acc[j] + bias;
            spbuf[(rt * 16 + m) * 65 + c0 + lm] = hv > 0.f ? hv : 0.f;   // hh, stride 65
        }
    }
    __syncthreads();

    // ---- Phase 4: LayerNorm over 64 channels per (b,t) ----
    if (tid < BT) {
        float s = 0.f;
        for (int o = 0; o < CC; ++o) s += spbuf[tid * 65 + o];
        float m = s * (1.0f / CC);
        float vv = 0.f;
        for (int o = 0; o < CC; ++o) { float d = spbuf[tid * 65 + o] - m; vv += d * d; }
        vv *= (1.0f / CC);
        mu[tid] = m;
        rs[tid] = rsqrtf(vv + LNEPS);
    }
    __syncthreads();

    for (int idx = tid; idx < BT * CC; idx += 128) {
        int bt = idx >> 6;
        int o  = idx & 63;
        int b  = bt / Tt;
        int t  = bt - b * Tt;
        float val = (spbuf[bt * 65 + o] - mu[bt]) * rs[bt] * gamma[o] + beta[o];
        out[((b * Nn + n) * CC + o) * Tt + t] = val;
    }
}

// ---------------- launch ----------------

extern "C" void kernel_launch(void* const* d_in, const int* in_sizes, int n_in,
                              void* d_out, int out_size, void* d_ws, size_t ws_size,
                              hipStream_t stream) {
    (void)in_sizes; (void)n_in; (void)out_size; (void)ws_size;
    const float* x     = (const float*)d_in[0];
    const int*   ei    = (const int*)  d_in[1];
    const float* lam   = (const float*)d_in[2];
    const float* chebW = (const float*)d_in[3];
    const float* chebB = (const float*)d_in[4];
    const float* timeW = (const float*)d_in[5];
    const float* timeB = (const float*)d_in[6];
    const float* resW  = (const float*)d_in[7];
    const float* resB  = (const float*)d_in[8];
    const float* gam   = (const float*)d_in[9];
    const float* bet   = (const float*)d_in[10];
    float* out = (float*)d_out;

    char* ws = (char*)d_ws;
    float* deg = (float*)ws;                                     // N floats
    float* v   = (float*)(ws + 40960);                           // TOTV floats
    float* t1v = (float*)(ws + 40960 + (size_t)TOTV * 4);        // TOTV floats
    float* agg = (float*)(ws + 40960 + (size_t)TOTV * 8);        // TOTV floats (also Tx2)

    const int TPB = 256;
    fill0<<<(Nn + TPB - 1) / TPB, TPB, 0, stream>>>(deg, Nn);
    fill0<<<(TOTV + TPB - 1) / TPB, TPB, 0, stream>>>(agg, TOTV);
    build_v<<<(TOTV + TPB - 1) / TPB, TPB, 0, stream>>>(x, v);
    deg_count<<<(En + TPB - 1) / TPB, TPB, 0, stream>>>(ei, deg);

    scatter_edges<<<(En * RL + TPB - 1) / TPB, TPB, 0, stream>>>(v, ei, agg);
    combine1<<<(TOTV + TPB - 1) / TPB, TPB, 0, stream>>>(v, agg, deg, lam, t1v);

    fill0<<<(TOTV + TPB - 1) / TPB, TPB, 0, stream>>>(agg, TOTV);
    scatter_edges<<<(En * RL + TPB - 1) / TPB, TPB, 0, stream>>>(t1v, ei, agg);
    combine2<<<(TOTV + TPB - 1) / TPB, TPB, 0, stream>>>(v, t1v, deg, lam, agg);

    fused_tail<<<Nn, 128, 0, stream>>>(v, t1v, agg, chebW, chebB, x,
                                       timeW, timeB, resW, resB, gam, bet, out);
}